// SOM_85787676770973
// MI455X (gfx1250) — compile-verified
//
#include <hip/hip_runtime.h>
#include <hip/hip_bf16.h>

// SOM distance: out[b, g] = sum_d (W[g,d] - X[b,d])^2
//             = ||x_b||^2 + ||w_g||^2 - 2 * dot(x_b, w_g)
// Cross term via V_WMMA_F32_16X16X4_F32 (full fp32 precision, matrix pipe).
// B=1024, G=64*64=4096, D=64. Output 16.8MB -> HBM-store bound (~0.8us floor).

typedef __attribute__((ext_vector_type(2))) float v2f;
typedef __attribute__((ext_vector_type(8))) float v8f;

#define BATCH 1024
#define GRIDN 4096
#define DIM   64

// ---------------------------------------------------------------------------
// Pass 1: row norms. ws[0..4095] = ||w_g||^2 ; ws[4096..5119] = ||x_b||^2
// ---------------------------------------------------------------------------
__global__ __launch_bounds__(256)
void SOM_norms_kernel(const float* __restrict__ x,
                      const float* __restrict__ w,
                      float* __restrict__ ws) {
    int t = blockIdx.x * blockDim.x + threadIdx.x;
    if (t < GRIDN) {
        const float4* p = (const float4*)(w + (size_t)t * DIM);
        float s = 0.f;
#pragma unroll
        for (int i = 0; i < DIM / 4; ++i) {
            float4 v = p[i];
            s += v.x * v.x + v.y * v.y + v.z * v.z + v.w * v.w;
        }
        ws[t] = s;
    } else if (t < GRIDN + BATCH) {
        int m = t - GRIDN;
        const float4* p = (const float4*)(x + (size_t)m * DIM);
        float s = 0.f;
#pragma unroll
        for (int i = 0; i < DIM / 4; ++i) {
            float4 v = p[i];
            s += v.x * v.x + v.y * v.y + v.z * v.z + v.w * v.w;
        }
        ws[GRIDN + m] = s;
    }
}

// ---------------------------------------------------------------------------
// Pass 2: one wave32 computes a 16(M) x 64(N) output tile.
// K=64 as 16 steps of V_WMMA_F32_16X16X4_F32; A fragment reused across the
// 4 N-subtiles. A/B 16x4 f32 fragment layout (ISA 7.12.2): lane<16 holds
// K={0,1} of row (lane), lane>=16 holds K={2,3} -> one float2 load each.
// ---------------------------------------------------------------------------
__global__ __launch_bounds__(256)
void SOM_dist_kernel(const float* __restrict__ x,
                     const float* __restrict__ w,
                     const float* __restrict__ wnorm,   // 4096
                     const float* __restrict__ xnorm,   // 1024
                     float* __restrict__ out) {
    const int lane = threadIdx.x & 31;
    const int wid  = blockIdx.x * (blockDim.x >> 5) + (threadIdx.x >> 5);

    const int mtile = wid >> 6;        // 64 M-tiles of 16 rows
    const int ntile = wid & 63;        // 64 N-tiles of 64 cols
    const int m0 = mtile * 16;
    const int n0 = ntile * 64;

    const int lm   = lane & 15;        // row/col within fragment
    const int half = lane >> 4;        // 0: K 0-1, 1: K 2-3

    const float* xrow = x + (size_t)(m0 + lm) * DIM + 2 * half;
    const float* wrow = w + (size_t)(n0 + lm) * DIM + 2 * half;

    v8f acc0 = {}, acc1 = {}, acc2 = {}, acc3 = {};

#pragma unroll
    for (int kk = 0; kk < DIM / 4; ++kk) {
        const int ko = kk * 4;
        v2f a  = *(const v2f*)(xrow + ko);
        v2f b0 = *(const v2f*)(wrow + ko);
        v2f b1 = *(const v2f*)(wrow + 16 * DIM + ko);
        v2f b2 = *(const v2f*)(wrow + 32 * DIM + ko);
        v2f b3 = *(const v2f*)(wrow + 48 * DIM + ko);
        // D = A*B + C  (8 args: neg_a, A, neg_b, B, c_mod, C, reuse_a, reuse_b)
        acc0 = __builtin_amdgcn_wmma_f32_16x16x4_f32(false, a, false, b0,
                                                     (short)0, acc0, false, false);
        acc1 = __builtin_amdgcn_wmma_f32_16x16x4_f32(false, a, false, b1,
                                                     (short)0, acc1, false, false);
        acc2 = __builtin_amdgcn_wmma_f32_16x16x4_f32(false, a, false, b2,
                                                     (short)0, acc2, false, false);
        acc3 = __builtin_amdgcn_wmma_f32_16x16x4_f32(false, a, false, b3,
                                                     (short)0, acc3, false, false);
    }

    // Epilogue. C/D layout: VGPR r, lanes 0-15 -> M=r, N=lane;
    //                       lanes 16-31 -> M=r+8, N=lane-16.
    float xr[8];
#pragma unroll
    for (int r = 0; r < 8; ++r) xr[r] = xnorm[m0 + r + 8 * half];

    v8f acc[4] = {acc0, acc1, acc2, acc3};
#pragma unroll
    for (int t = 0; t < 4; ++t) {
        const int n  = n0 + t * 16 + lm;
        const float wn = wnorm[n];
#pragma unroll
        for (int r = 0; r < 8; ++r) {
            const int m = m0 + r + 8 * half;
            out[(size_t)m * GRIDN + n] = fmaf(-2.f, acc[t][r], xr[r] + wn);
        }
    }
}

extern "C" void kernel_launch(void* const* d_in, const int* in_sizes, int n_in,
                              void* d_out, int out_size, void* d_ws, size_t ws_size,
                              hipStream_t stream) {
    const float* x = (const float*)d_in[0];   // (1024, 64)
    const float* w = (const float*)d_in[1];   // (64, 64, 64) -> (4096, 64)
    float* out = (float*)d_out;               // (1024, 4096)
    float* ws  = (float*)d_ws;                // [0,4096): ||w||^2, [4096,5120): ||x||^2

    // Pass 1: norms (5120 threads)
    SOM_norms_kernel<<<(GRIDN + BATCH + 255) / 256, 256, 0, stream>>>(x, w, ws);

    // Pass 2: 4096 waves = 512 blocks x 256 threads (8 waves/block)
    SOM_dist_kernel<<<512, 256, 0, stream>>>(x, w, ws, ws + GRIDN, out);
}